// FlatDilation1D_9474697855597
// MI455X (gfx1250) — compile-verified
//
#include <hip/hip_runtime.h>
#include <hip/hip_bf16.h>
#include <math.h>

#define KSZ 41

typedef __attribute__((ext_vector_type(2))) float v2f;
typedef __attribute__((ext_vector_type(4))) float v4f;
typedef __attribute__((ext_vector_type(8))) float v8f;

// Phase 1: compute base[0..40] into d_ws.
// base[i] = max_{k : |k-20| <= scale} input[a + (i+k)%41]
__global__ void fd_base_kernel(const float* __restrict__ in,
                               const float* __restrict__ scale_p,
                               float* __restrict__ base,
                               long long a) {
    __shared__ float padded[KSZ];
    int t = threadIdx.x;
    if (t < KSZ) padded[t] = in[a + t];
    __syncthreads();
    if (t < KSZ) {
        float s = *scale_p;
        float m = -INFINITY;
#pragma unroll
        for (int k = 0; k < KSZ; ++k) {
            if (fabsf((float)k - 20.0f) <= s) {
                int idx = t + k;
                if (idx >= KSZ) idx -= KSZ;
                m = fmaxf(m, padded[idx]);
            }
        }
        base[t] = m;
    }
}

// Phase 2: out[t] = base[t % 41].
// numWaves is a multiple of 41, so tile % 41 (hence the tile's 256 values) is
// invariant per wave. Each wave builds its 256-value tile ONCE as a 16x16 f32
// WMMA D-matrix:
//   D = sum_{j=0..3} A_j x B_j,  A_j[m,k] = (m == 4j+k),
//   B_j[k,n] = base[(cb + (4j+k)*16 + n) % 41]
// => D[m,n] = base[(cb + m*16 + n) % 41], exact (one-hot f32 select).
// The tile is then permuted through LDS into lane-contiguous float4s and the
// hot loop is nothing but two global_store_b128 per wave per 1KB tile.
__global__ void __launch_bounds__(256)
fd_broadcast_kernel(const float* __restrict__ base,
                    float* __restrict__ out,
                    long long n) {
    __shared__ float sB[KSZ];
    __shared__ __align__(16) float sTile[8][256];   // one 1KB tile per wave

    if (threadIdx.x < KSZ) sB[threadIdx.x] = base[threadIdx.x];
    __syncthreads();

    const int  lane   = threadIdx.x & 31;
    const bool hi     = lane >= 16;
    const int  ncol   = hi ? lane - 16 : lane;  // matrix column held by this lane
    const int  rowAdd = hi ? 8 : 0;             // C/D layout: vgpr r -> row r / r+8
    const int  wIB    = threadIdx.x >> 5;       // wave index within block

    const long long wave     = (long long)blockIdx.x * (blockDim.x >> 5) + wIB;
    const long long numWaves = (long long)gridDim.x * (blockDim.x >> 5); // multiple of 41
    const long long numTiles = n >> 8;          // 256 outputs per tile

    // Tile content for this wave (invariant across its tiles).
    const int cbm = (int)((wave << 8) % KSZ);
    v8f d = {};
#if __has_builtin(__builtin_amdgcn_wmma_f32_16x16x4_f32)
    // One-hot A matrices. A layout (16x4 f32): lanes 0-15 hold (M=lane, K=0/1),
    // lanes 16-31 hold (M=lane-16, K=2/3).
    {
        const int kb = hi ? 2 : 0;
#pragma unroll
        for (int j = 0; j < 4; ++j) {
            v2f A;
            A.x = (ncol == 4 * j + kb)     ? 1.0f : 0.0f;
            A.y = (ncol == 4 * j + kb + 1) ? 1.0f : 0.0f;
            // B layout (4x16 f32), mirroring the C/D row split:
            // VGPR0: lanes0-15 row K=0, lanes16-31 row K=2; VGPR1: K=1 / K=3.
            const int mb = 4 * j + kb;
            v2f B;
            B.x = sB[(cbm + mb * 16 + ncol) % KSZ];
            B.y = sB[(cbm + (mb + 1) * 16 + ncol) % KSZ];
            d = __builtin_amdgcn_wmma_f32_16x16x4_f32(
                    /*neg_a=*/false, A, /*neg_b=*/false, B,
                    /*c_mod=*/(short)0, d, /*reuse_a=*/false, /*reuse_b=*/false);
        }
    }
#else
    // Fallback: gather directly into the D register layout (identical result).
#pragma unroll
    for (int r = 0; r < 8; ++r)
        d[r] = sB[(cbm + (r + rowAdd) * 16 + ncol) % KSZ];
#endif

    // Permute D (column-strided per lane) -> lane-contiguous float4s via LDS.
    float* tw = sTile[wIB];
#pragma unroll
    for (int r = 0; r < 8; ++r)
        tw[(r + rowAdd) * 16 + ncol] = d[r];
    __syncthreads();

    const v4f lo = ((const v4f*)tw)[lane];        // floats [lane*4 .. lane*4+3]
    const v4f hi4 = ((const v4f*)tw)[lane + 32];  // floats [512B half + lane*4 ..]

    // Hot loop: pure b128 store stream. Each instruction writes a fully
    // contiguous 512B run per wave; two instructions cover the 1KB tile.
    const long long strideF = numWaves << 8;      // floats between this wave's tiles
    float* p = out + (wave << 8);
    for (long long tile = wave; tile < numTiles; tile += numWaves) {
        ((v4f*)p)[lane]      = lo;
        ((v4f*)p)[lane + 32] = hi4;
        p += strideF;
    }

    // Tail (n mod 256 elements) — scalar, no WMMA involved.
    if (blockIdx.x == 0 && threadIdx.x == 0) {
        for (long long t = numTiles << 8; t < n; ++t)
            out[t] = sB[(int)(t % KSZ)];
    }
}

extern "C" void kernel_launch(void* const* d_in, const int* in_sizes, int n_in,
                              void* d_out, int out_size, void* d_ws, size_t ws_size,
                              hipStream_t stream) {
    const float* x       = (const float*)d_in[0];
    const float* scale_p = (const float*)d_in[1];
    float*       out     = (float*)d_out;
    float*       base    = (float*)d_ws;   // 41 floats of scratch

    const long long n = (long long)out_size;  // == in_sizes[0]
    // a = -((41 - n) // 2 + 2), Python floor division (arithmetic >> floors).
    const long long missing = (long long)KSZ - n;
    const long long a = -((missing >> 1) + 2);

    fd_base_kernel<<<1, 64, 0, stream>>>(x, scale_p, base, a);

    // 656 blocks * 8 waves = 5248 waves = 41 * 128: numWaves % 41 == 0 keeps
    // each wave's tile residue (tile % 41) constant across its grid-stride.
    fd_broadcast_kernel<<<656, 256, 0, stream>>>(base, out, n);
}